// GCPNetPredictor_26929444946404
// MI455X (gfx1250) — compile-verified
//
#include <hip/hip_runtime.h>
#include <hip/hip_bf16.h>

typedef __attribute__((ext_vector_type(2))) float v2f;
typedef __attribute__((ext_vector_type(4))) float v4f;
typedef __attribute__((ext_vector_type(8))) float v8f;

#define BQ 16
#define LQ 1024
#define DQ 256
#define KQ 16
#define NN (BQ * LQ)          /* 16384 nodes  */
#define NEDGE (NN * KQ)       /* 262144 edges */

/* output layout (flat concat, return order: e, chi, xi, x_bb) */
#define OFF_E   0ull
#define OFF_CHI ((unsigned long long)NEDGE * 2ull * DQ)          /* 134217728 */
#define OFF_XI  (OFF_CHI + (unsigned long long)NN * 6ull)        /* +98304    */
#define OFF_XBB (OFF_XI + (unsigned long long)NEDGE * 3ull)      /* +786432   */

#define WSTRIDE 258   /* even (8B-aligned b64 reads), 258%64==2 -> conflict-free rows */

/* ---------------- Kernel 1: x_bb = h @ W^T via V_WMMA_F32_16X16X4_F32 --------
 * 4 waves/block, each wave owns one 16-row tile; W staged zero-padded in LDS
 * so the 64-step K loop is branchless: global_load_b64 (A) + ds_load_b64 (B)
 * + v_wmma_f32_16x16x4_f32.                                                   */
__global__ __launch_bounds__(128) void k_gemm_xbb(const float* __restrict__ h,
                                                  const float* __restrict__ W,
                                                  float* __restrict__ xbb) {
    __shared__ float Wl[16 * WSTRIDE];
    for (int idx = threadIdx.x; idx < 16 * 256; idx += 128) {
        const int r = idx >> 8, c = idx & 255;
        Wl[r * WSTRIDE + c] = (r < 9) ? W[r * DQ + c] : 0.f;
    }
    __syncthreads();

    const int wave = threadIdx.x >> 5;
    const int l    = threadIdx.x & 31;
    const int tile = blockIdx.x * 4 + wave;     // 0..1023
    const int row0 = tile * 16;
    const int m    = l & 15;                    // A row / B col
    const int kh   = l >> 4;                    // K half: lanes 0-15 K=0,1; 16-31 K=2,3

    const float* arow = h + (size_t)(row0 + m) * DQ + 2 * kh;
    const float* brow = &Wl[m * WSTRIDE + 2 * kh];

    v8f acc = {};
    for (int k0 = 0; k0 < DQ; k0 += 4) {
        v2f a = *(const v2f*)(arow + k0);
        v2f b = *(const v2f*)(brow + k0);
        acc = __builtin_amdgcn_wmma_f32_16x16x4_f32(
            false, a, false, b, (short)0, acc, false, false);
    }
    /* C layout: VGPR v, lane l -> M = v + 8*(l>>4), N = l&15 */
    if (m < 9) {
        #pragma unroll
        for (int v = 0; v < 8; ++v) {
            const int M = v + 8 * kh;
            xbb[(size_t)(row0 + M) * 9 + m] = acc[v];
        }
    }
}

/* ---------------- Kernel 2: xs (padded float4, w=0) + squared norms ---------- */
__global__ void k_prep(const float* __restrict__ xbb,
                       v4f* __restrict__ xs4,
                       float* __restrict__ norms) {
    const int g = blockIdx.x * blockDim.x + threadIdx.x;
    if (g >= NN) return;
    const float x = xbb[(size_t)g * 9 + 3];
    const float y = xbb[(size_t)g * 9 + 4];
    const float z = xbb[(size_t)g * 9 + 5];
    v4f p; p.x = x; p.y = y; p.z = z; p.w = 0.f;
    xs4[g] = p;
    norms[g] = x * x + y * y + z * z;
}

/* ---------------- Kernel 3: pairwise d2 via WMMA + stable top-16 -------------
 * one wave per 16-row tile; dot tile -> LDS; lanes 0-15 keep sorted lists.    */
__global__ __launch_bounds__(32) void k_topk(const v4f* __restrict__ xs4,
                                             const float* __restrict__ norms,
                                             int* __restrict__ srcIdx) {
    __shared__ float tile[16 * 16];
    __shared__ float listD[16 * 16];
    __shared__ int   listI[16 * 16];

    const int t     = blockIdx.x;       // 0..1023
    const int b     = t >> 6;           // batch
    const int row0  = (t & 63) << 4;    // row tile within batch
    const int gbase = b * LQ;
    const int l  = threadIdx.x;
    const int m  = l & 15;
    const int kh = l >> 4;

    /* A fragment (loop-invariant): components (2kh, 2kh+1) of xs row */
    const v4f xr = xs4[gbase + row0 + m];
    v2f a;
    a.x = kh ? xr.z : xr.x;
    a.y = kh ? xr.w : xr.y;
    const float ni = norms[gbase + row0 + m];

    if (l < 16) {
        for (int k = 0; k < KQ; ++k) { listD[l * 16 + k] = 3.0e38f; listI[l * 16 + k] = 0; }
    }
    __syncthreads();

    for (int j0 = 0; j0 < LQ; j0 += 16) {
        /* B fragment: B[k][n] = xs[j0+n][k] */
        const v4f xc = xs4[gbase + j0 + m];
        v2f bb;
        bb.x = kh ? xc.z : xc.x;
        bb.y = kh ? xc.w : xc.y;
        v8f acc = {};
        acc = __builtin_amdgcn_wmma_f32_16x16x4_f32(
            false, a, false, bb, (short)0, acc, false, false);
        #pragma unroll
        for (int v = 0; v < 8; ++v)
            tile[(v + 8 * kh) * 16 + m] = acc[v];
        __syncthreads();

        if (l < 16) {
            const int i = row0 + l;
            for (int c = 0; c < 16; ++c) {
                const int j = j0 + c;
                if (j == i) continue;    /* diagonal: the +1e9 entry never ranks */
                const float d = ni + norms[gbase + j] - 2.0f * tile[l * 16 + c];
                if (d < listD[l * 16 + 15]) {
                    int p = 15;
                    while (p > 0 && listD[l * 16 + p - 1] > d) {   /* strict: stable */
                        listD[l * 16 + p] = listD[l * 16 + p - 1];
                        listI[l * 16 + p] = listI[l * 16 + p - 1];
                        --p;
                    }
                    listD[l * 16 + p] = d;
                    listI[l * 16 + p] = j;
                }
            }
        }
        __syncthreads();
    }

    if (l < 16) {
        const int i = gbase + row0 + l;   /* global node */
        for (int k = 0; k < KQ; ++k)
            srcIdx[(size_t)i * KQ + k] = gbase + listI[l * 16 + k];
    }
}

/* ---------------- Kernel 4: e = concat(h[src], h[tgt]) — bandwidth king ------
 * one block per edge; 128 threads x float4 = 512 floats. NT stores keep h
 * resident in the 192MB L2 while streaming the 512MB write-once output.      */
__global__ __launch_bounds__(128) void k_edges(const v4f* __restrict__ h4,
                                               const int* __restrict__ srcIdx,
                                               v4f* __restrict__ e4) {
    const long long eid = (long long)blockIdx.x;
    const int t = threadIdx.x;                  // 0..127
    const int src = srcIdx[eid];
    const int tgt = (int)(eid >> 4);            // eid / K = global target node
    const int node = (t < (DQ / 4)) ? src : tgt;
    const int c = t & (DQ / 4 - 1);             // 0..63
    const v4f val = h4[(size_t)node * (DQ / 4) + c];
    __builtin_nontemporal_store(val, &e4[eid * 128ll + t]);
}

/* ---------------- Kernel 5: xi = normalize(xs[src] - xs[tgt]) ---------------- */
__global__ void k_xi(const v4f* __restrict__ xs4,
                     const int* __restrict__ srcIdx,
                     float* __restrict__ xi) {
    const int eid = blockIdx.x * blockDim.x + threadIdx.x;
    if (eid >= NEDGE) return;
    const int src = srcIdx[eid];
    const int tgt = eid >> 4;
    const v4f pa = xs4[src];
    const v4f pc = xs4[tgt];
    const float vx = pa.x - pc.x, vy = pa.y - pc.y, vz = pa.z - pc.z;
    const float n = sqrtf(vx * vx + vy * vy + vz * vz);
    const float inv = 1.0f / fmaxf(n, 1e-12f);
    __builtin_nontemporal_store(vx * inv, &xi[(size_t)eid * 3 + 0]);
    __builtin_nontemporal_store(vy * inv, &xi[(size_t)eid * 3 + 1]);
    __builtin_nontemporal_store(vz * inv, &xi[(size_t)eid * 3 + 2]);
}

/* ---------------- Kernel 6: chi = [fwdv, bwdv] ------------------------------- */
__global__ void k_chi(const v4f* __restrict__ xs4,
                      float* __restrict__ chi) {
    const int g = blockIdx.x * blockDim.x + threadIdx.x;
    if (g >= NN) return;
    const int lpos = g & (LQ - 1);
    const v4f p = xs4[g];
    float fx = 0.f, fy = 0.f, fz = 0.f, bx = 0.f, by = 0.f, bz = 0.f;
    if (lpos < LQ - 1) {
        const v4f q = xs4[g + 1];
        const float vx = q.x - p.x, vy = q.y - p.y, vz = q.z - p.z;
        const float inv = 1.0f / fmaxf(sqrtf(vx * vx + vy * vy + vz * vz), 1e-12f);
        fx = vx * inv; fy = vy * inv; fz = vz * inv;
    }
    if (lpos > 0) {
        const v4f q = xs4[g - 1];
        const float vx = q.x - p.x, vy = q.y - p.y, vz = q.z - p.z;
        const float inv = 1.0f / fmaxf(sqrtf(vx * vx + vy * vy + vz * vz), 1e-12f);
        bx = vx * inv; by = vy * inv; bz = vz * inv;
    }
    float* c = chi + (size_t)g * 6;
    c[0] = fx; c[1] = fy; c[2] = fz;
    c[3] = bx; c[4] = by; c[5] = bz;
}

extern "C" void kernel_launch(void* const* d_in, const int* in_sizes, int n_in,
                              void* d_out, int out_size, void* d_ws, size_t ws_size,
                              hipStream_t stream) {
    const float* feats = (const float*)d_in[0];   /* (B,L,D) f32 */
    /* d_in[1] mask, d_in[3] batch_indices, d_in[4] x_slice_index: unused */
    const float* W = (const float*)d_in[2];       /* (9,D) f32 */

    float* out  = (float*)d_out;
    float* e    = out + OFF_E;
    float* chi  = out + OFF_CHI;
    float* xi   = out + OFF_XI;
    float* xbb  = out + OFF_XBB;

    char* ws = (char*)d_ws;
    v4f*   xs4   = (v4f*)ws;                                  /* NN*16 B  */
    float* norms = (float*)(ws + (size_t)NN * 16);            /* NN*4  B  */
    int*   srcIx = (int*)(ws + (size_t)NN * 16 + NN * 4);     /* NEDGE*4 B */

    k_gemm_xbb<<<NN / 64, 128, 0, stream>>>(feats, W, xbb);
    k_prep<<<(NN + 255) / 256, 256, 0, stream>>>(xbb, xs4, norms);
    k_topk<<<NN / 16, 32, 0, stream>>>(xs4, norms, srcIx);
    k_edges<<<NEDGE, 128, 0, stream>>>((const v4f*)feats, srcIx, (v4f*)e);
    k_xi<<<(NEDGE + 255) / 256, 256, 0, stream>>>(xs4, srcIx, xi);
    k_chi<<<(NN + 255) / 256, 256, 0, stream>>>(xs4, chi);
}